// GATModel_86681029968550
// MI455X (gfx1250) — compile-verified
//
#include <hip/hip_runtime.h>
#include <math.h>

#define HIDC 32
#define NHEADS 4
#define FU 128
#define FP 64

typedef float v2f __attribute__((ext_vector_type(2)));
typedef float v8f __attribute__((ext_vector_type(8)));

// ---------------- utility atomics ----------------
__device__ __forceinline__ void atomic_max_f32(float* addr, float val) {
  int* ia = reinterpret_cast<int*>(addr);
  int old = __float_as_int(*addr);
  while (__int_as_float(old) < val) {
    int assumed = old;
    old = atomicCAS(ia, assumed, __float_as_int(val));
    if (old == assumed) break;
  }
}

__global__ void fill_f32v4(float4* __restrict__ p, float v, long long n4) {
  long long i = (long long)blockIdx.x * blockDim.x + threadIdx.x;
  if (i < n4) p[i] = make_float4(v, v, v, v);
}

// ---------------- fp32 WMMA GEMM: C[M,Nc] = A[M,K] * B[K,Nc] ----------------
// blockDim = (32, numColTiles). One wave computes one 16x16 tile via
// V_WMMA_F32_16X16X4_F32, stepping K by 4. M multiple of 16, Nc multiple of
// 16*blockDim.y, K multiple of 4. EXEC stays all-ones (no divergence).
__global__ void gemm_wmma_f32(const float* __restrict__ A, const float* __restrict__ B,
                              float* __restrict__ C, int K, int Nc) {
  const int lane = threadIdx.x;                 // 0..31 (wave32)
  const int rowTile = blockIdx.x;
  const int colTile = threadIdx.y;              // one wave per col tile
  const int row = rowTile * 16 + (lane & 15);
  const int col = colTile * 16 + (lane & 15);
  const int kb = (lane >> 4) * 2;               // lanes 16-31 hold K=2,3 of each step

  v8f acc = {0.f, 0.f, 0.f, 0.f, 0.f, 0.f, 0.f, 0.f};
  const float* ap = A + (long long)row * K + kb;
  const float* bp = B + (long long)kb * Nc + col;
  for (int k0 = 0; k0 < K; k0 += 4) {
    v2f a, b;
    a.x = ap[k0];                               // A[row, k0+kb]
    a.y = ap[k0 + 1];                           // A[row, k0+kb+1]
    b.x = bp[(long long)k0 * Nc];               // B[k0+kb,   col]
    b.y = bp[(long long)(k0 + 1) * Nc];         // B[k0+kb+1, col]
    acc = __builtin_amdgcn_wmma_f32_16x16x4_f32(false, a, false, b, (short)0, acc,
                                                false, false);
  }
  // D layout (32-bit 16x16): VGPR v -> M = v + 8*(lane>=16), N = lane&15
  const int mbase = rowTile * 16 + ((lane >> 4) << 3);
#pragma unroll
  for (int v = 0; v < 8; ++v)
    C[(long long)(mbase + v) * Nc + col] = acc[v];
}

// ---------------- attention coefficients: as[n,h]=<h[n,h,:],a_s[h,:]> ----------------
__global__ void attn_coeff(const float* __restrict__ h, const float* __restrict__ a_s,
                           const float* __restrict__ a_d, float* __restrict__ s_out,
                           float* __restrict__ d_out_, int N, int H) {
  long long t = (long long)blockIdx.x * blockDim.x + threadIdx.x;
  if (t >= (long long)N * H) return;
  int hh = (int)(t % H);
  long long n = t / H;
  const float* hp = h + (n * H + hh) * HIDC;
  const float* asw = a_s + hh * HIDC;
  const float* adw = a_d + hh * HIDC;
  float ss = 0.f, sd = 0.f;
#pragma unroll
  for (int j = 0; j < HIDC; ++j) {
    float x = hp[j];
    ss += x * asw[j];
    sd += x * adw[j];
  }
  s_out[t] = ss;
  d_out_[t] = sd;
}

// ---------------- edge pass 1: leaky-relu logits + segment max ----------------
// One thread per edge; H compile-time so the per-head loop fully unrolls and
// src/dst are fetched once per edge.
template <int H>
__global__ void edge_logits(const int* __restrict__ ei, int E, long long Etot,
                            const float* __restrict__ as, const float* __restrict__ ad,
                            float* __restrict__ ebuf, float* __restrict__ mbuf) {
  long long k = (long long)blockIdx.x * blockDim.x + threadIdx.x;
  if (k >= Etot) return;
  int s, d;
  if (k < E) { s = ei[k]; d = ei[(long long)E + k]; }
  else       { s = d = (int)(k - E); }            // self loops
  const float* asp = as + (long long)s * H;
  const float* adp = ad + (long long)d * H;
  float* mp = mbuf + (long long)d * H;
  float* ep = ebuf + k * H;
#pragma unroll
  for (int h = 0; h < H; ++h) {
    float e = asp[h] + adp[h];
    e = (e > 0.f) ? e : 0.2f * e;
    ep[h] = e;
    atomic_max_f32(&mp[h], e);
  }
}

// ---------------- edge pass 2: exp(e - m[dst]) + segment sum ----------------
template <int H>
__global__ void edge_exp(const int* __restrict__ ei, int E, long long Etot,
                         const float* __restrict__ mbuf, float* __restrict__ ebuf,
                         float* __restrict__ den) {
  long long k = (long long)blockIdx.x * blockDim.x + threadIdx.x;
  if (k >= Etot) return;
  int d;
  if (k < E) { d = ei[(long long)E + k]; }
  else       { d = (int)(k - E); }
  const float* mp = mbuf + (long long)d * H;
  float* dp = den + (long long)d * H;
  float* ep = ebuf + k * H;
#pragma unroll
  for (int h = 0; h < H; ++h) {
    float ex = __expf(ep[h] - mp[h]);
    ep[h] = ex;
    unsafeAtomicAdd(&dp[h], ex);
  }
}

// ---------------- edge pass 3: out[dst] += h[src] * alpha (float4) ----------------
// One thread per 4 channels of one edge: b128 gather + 4 hw float atomics.
// For C=128 a full wave covers one contiguous 512B source row.
template <int H, int C>
__global__ void edge_scatter(const int* __restrict__ ei, int E, long long Etot,
                             const float* __restrict__ hsrc, const float* __restrict__ ebuf,
                             const float* __restrict__ den, float* __restrict__ out) {
  constexpr int CV = C / 4;
  long long t = (long long)blockIdx.x * blockDim.x + threadIdx.x;
  if (t >= Etot * CV) return;
  int c4 = (int)(t % CV);
  long long k = t / CV;
  int s, d;
  if (k < E) { s = ei[k]; d = ei[(long long)E + k]; }
  else       { s = d = (int)(k - E); }
  const int c = c4 * 4;
  const int hh = c / HIDC;  // head block of this channel group (4 | 32)
  float alpha = ebuf[k * H + hh] / den[(long long)d * H + hh];
  const float4 hv = *reinterpret_cast<const float4*>(hsrc + (long long)s * C + c);
  float* op = out + (long long)d * C + c;
  unsafeAtomicAdd(op + 0, hv.x * alpha);
  unsafeAtomicAdd(op + 1, hv.y * alpha);
  unsafeAtomicAdd(op + 2, hv.z * alpha);
  unsafeAtomicAdd(op + 3, hv.w * alpha);
}

// ---------------- x = relu(x + b), vectorized float4 ----------------
__global__ void bias_relu_v4(float4* __restrict__ x, const float4* __restrict__ b,
                             long long n4, int C4) {
  long long t = (long long)blockIdx.x * blockDim.x + threadIdx.x;
  if (t >= n4) return;
  float4 v = x[t];
  float4 bb = b[(int)(t % C4)];
  v.x += bb.x; v.y += bb.y; v.z += bb.z; v.w += bb.w;
  v.x = v.x > 0.f ? v.x : 0.f;
  v.y = v.y > 0.f ? v.y : 0.f;
  v.z = v.z > 0.f ? v.z : 0.f;
  v.w = v.w > 0.f ? v.w : 0.f;
  x[t] = v;
}

// ---------------- FC head: sigmoid(relu([x3,post]@w1+b1)@w2+b2) ----------------
__global__ void fc_head(const float* __restrict__ x3, const float* __restrict__ post,
                        const float* __restrict__ w1, const float* __restrict__ b1,
                        const float* __restrict__ w2, const float* __restrict__ b2,
                        float* __restrict__ out, int N) {
  int n = blockIdx.x * blockDim.x + threadIdx.x;
  if (n >= N) return;
  const float* xr = x3 + (long long)n * HIDC;
  const float* pr = post + (long long)n * FP;
  float hid[HIDC];
  for (int o = 0; o < HIDC; ++o) {
    float s = b1[o];
#pragma unroll 8
    for (int j = 0; j < HIDC; ++j) s += xr[j] * w1[j * HIDC + o];
#pragma unroll 8
    for (int j = 0; j < FP; ++j) s += pr[j] * w1[(HIDC + j) * HIDC + o];
    hid[o] = s > 0.f ? s : 0.f;
  }
  float z = b2[0];
#pragma unroll
  for (int o = 0; o < HIDC; ++o) z += hid[o] * w2[o];
  out[n] = 1.f / (1.f + __expf(-z));
}

extern "C" void kernel_launch(void* const* d_in, const int* in_sizes, int n_in,
                              void* d_out, int out_size, void* d_ws, size_t ws_size,
                              hipStream_t stream) {
  const float* user = (const float*)d_in[0];
  const float* post = (const float*)d_in[1];
  const int*   ei   = (const int*)d_in[2];    // [2,E] src row then dst row
  const float* W1   = (const float*)d_in[3];
  const float* a1s  = (const float*)d_in[4];
  const float* a1d  = (const float*)d_in[5];
  const float* b1   = (const float*)d_in[6];
  const float* W2   = (const float*)d_in[7];
  const float* a2s  = (const float*)d_in[8];
  const float* a2d  = (const float*)d_in[9];
  const float* b2   = (const float*)d_in[10];
  const float* fc1w = (const float*)d_in[11];
  const float* fc1b = (const float*)d_in[12];
  const float* fc2w = (const float*)d_in[13];
  const float* fc2b = (const float*)d_in[14];

  const int N = in_sizes[0] / FU;       // 100000 (multiple of 16)
  const int E = in_sizes[2] / 2;        // 1600000
  const long long Etot = (long long)E + N;
  const int C1 = NHEADS * HIDC;         // 128

  float* ws = (float*)d_ws;
  long long off = 0;
  float* h1  = ws + off; off += (long long)N * C1;
  float* x2  = ws + off; off += (long long)N * C1;
  float* h2  = ws + off; off += (long long)N * HIDC;
  float* x3  = ws + off; off += (long long)N * HIDC;
  float* as1 = ws + off; off += (long long)N * NHEADS;
  float* ad1 = ws + off; off += (long long)N * NHEADS;
  float* m1  = ws + off; off += (long long)N * NHEADS;
  float* dn1 = ws + off; off += (long long)N * NHEADS;
  float* as2 = ws + off; off += N;
  float* ad2 = ws + off; off += N;
  float* m2  = ws + off; off += N;
  float* dn2 = ws + off; off += N;
  float* e1  = ws + off; off += Etot * NHEADS;
  float* e2  = ws + off; off += Etot;

  auto nb = [](long long n, int bs) { return (unsigned)((n + bs - 1) / bs); };

  // ---------------- GAT layer 1 ----------------
  gemm_wmma_f32<<<N / 16, dim3(32, 8), 0, stream>>>(user, W1, h1, FU, C1);
  attn_coeff<<<nb((long long)N * NHEADS, 256), 256, 0, stream>>>(h1, a1s, a1d, as1, ad1, N, NHEADS);
  fill_f32v4<<<nb((long long)N * NHEADS / 4, 256), 256, 0, stream>>>((float4*)m1, -3.0e38f, (long long)N * NHEADS / 4);
  fill_f32v4<<<nb((long long)N * NHEADS / 4, 256), 256, 0, stream>>>((float4*)dn1, 0.f, (long long)N * NHEADS / 4);
  fill_f32v4<<<nb((long long)N * C1 / 4, 256), 256, 0, stream>>>((float4*)x2, 0.f, (long long)N * C1 / 4);
  edge_logits<NHEADS><<<nb(Etot, 256), 256, 0, stream>>>(ei, E, Etot, as1, ad1, e1, m1);
  edge_exp<NHEADS><<<nb(Etot, 256), 256, 0, stream>>>(ei, E, Etot, m1, e1, dn1);
  edge_scatter<NHEADS, 128><<<nb(Etot * (128 / 4), 256), 256, 0, stream>>>(ei, E, Etot, h1, e1, dn1, x2);
  bias_relu_v4<<<nb((long long)N * C1 / 4, 256), 256, 0, stream>>>((float4*)x2, (const float4*)b1,
                                                                   (long long)N * C1 / 4, C1 / 4);

  // ---------------- GAT layer 2 (1 head) ----------------
  gemm_wmma_f32<<<N / 16, dim3(32, 2), 0, stream>>>(x2, W2, h2, C1, HIDC);
  attn_coeff<<<nb(N, 256), 256, 0, stream>>>(h2, a2s, a2d, as2, ad2, N, 1);
  fill_f32v4<<<nb(N / 4, 256), 256, 0, stream>>>((float4*)m2, -3.0e38f, N / 4);
  fill_f32v4<<<nb(N / 4, 256), 256, 0, stream>>>((float4*)dn2, 0.f, N / 4);
  fill_f32v4<<<nb((long long)N * HIDC / 4, 256), 256, 0, stream>>>((float4*)x3, 0.f, (long long)N * HIDC / 4);
  edge_logits<1><<<nb(Etot, 256), 256, 0, stream>>>(ei, E, Etot, as2, ad2, e2, m2);
  edge_exp<1><<<nb(Etot, 256), 256, 0, stream>>>(ei, E, Etot, m2, e2, dn2);
  edge_scatter<1, HIDC><<<nb(Etot * (HIDC / 4), 256), 256, 0, stream>>>(ei, E, Etot, h2, e2, dn2, x3);
  bias_relu_v4<<<nb((long long)N * HIDC / 4, 256), 256, 0, stream>>>((float4*)x3, (const float4*)b2,
                                                                     (long long)N * HIDC / 4, HIDC / 4);

  // ---------------- FC head ----------------
  fc_head<<<nb(N, 256), 256, 0, stream>>>(x3, post, fc1w, fc1b, fc2w, fc2b, (float*)d_out, N);
}